// WorkingMemory_35699768165146
// MI455X (gfx1250) — compile-verified
//
#include <hip/hip_runtime.h>
#include <math.h>

#if defined(__HIP_DEVICE_COMPILE__)
#if !__has_builtin(__builtin_amdgcn_wmma_f32_16x16x4_f32)
#error "gfx1250 wmma f32 16x16x4 builtin not available"
#endif
#endif

typedef __attribute__((ext_vector_type(2))) float v2f;
typedef __attribute__((ext_vector_type(8))) float v8f;

static constexpr int kD = 512;    // embed dim
static constexpr int kB = 8192;   // batch == memory size (buf == input_data)

static __device__ __forceinline__ v8f wmma4(v2f a, v2f b, v8f c) {
  // D = A(16x4) * B(4x16) + C(16x16), fp32 exact
  return __builtin_amdgcn_wmma_f32_16x16x4_f32(false, a, false, b, (short)0, c,
                                               false, false);
}

// ---------------------------------------------------------------------------
// C[M][N] = A[MxK] * B[NxK]^T + bias[N]   (row-major, NT)
// gateMode: in-place epilogue C = sigmoid(acc+bias)*C + (1-sigmoid)*Xres
// Block: 256 threads = 8 waves, block tile 32(M) x 64(N), K staged in LDS.
// ---------------------------------------------------------------------------
__global__ __launch_bounds__(256)
void gemm_nt_bias(const float* __restrict__ A, const float* __restrict__ Bm,
                  const float* __restrict__ bias, float* __restrict__ Cm,
                  int Ndim, int Kdim, int gateMode,
                  const float* __restrict__ Xres) {
  constexpr int LS = 68;  // LDS row stride (floats): 16B-aligned, bank-skewed
  __shared__ float As[32 * LS];
  __shared__ float Bs[64 * LS];

  const int tid  = threadIdx.x;
  const int lane = tid & 31;
  const int wave = tid >> 5;
  const int ml = lane & 15;   // N (or M) index within tile
  const int kh = lane >> 4;   // lane-half selects K pair / row group
  const int waveM = wave >> 2, waveN = wave & 3;
  const int M0 = blockIdx.y * 32, N0 = blockIdx.x * 64;

  v8f acc = {0.f, 0.f, 0.f, 0.f, 0.f, 0.f, 0.f, 0.f};

  for (int k0 = 0; k0 < Kdim; k0 += 64) {
    // stage A 32x64 and B 64x64 (float4, coalesced)
    for (int i = tid; i < 32 * 16; i += 256) {
      int r = i >> 4, c = (i & 15) << 2;
      *(float4*)&As[r * LS + c] =
          *(const float4*)&A[(size_t)(M0 + r) * Kdim + k0 + c];
    }
    for (int i = tid; i < 64 * 16; i += 256) {
      int r = i >> 4, c = (i & 15) << 2;
      *(float4*)&Bs[r * LS + c] =
          *(const float4*)&Bm[(size_t)(N0 + r) * Kdim + k0 + c];
    }
    __syncthreads();
#pragma unroll
    for (int kk = 0; kk < 64; kk += 4) {
      v2f a = *(const v2f*)&As[(waveM * 16 + ml) * LS + kk + kh * 2];
      v2f b = *(const v2f*)&Bs[(waveN * 16 + ml) * LS + kk + kh * 2];
      acc = wmma4(a, b, acc);
    }
    __syncthreads();
  }

  const int col = N0 + waveN * 16 + ml;
  const float bcol = bias[col];
#pragma unroll
  for (int r = 0; r < 8; ++r) {
    int row = M0 + waveM * 16 + kh * 8 + r;
    size_t idx = (size_t)row * Ndim + col;
    float v = acc[r] + bcol;
    if (gateMode) {
      float g = 1.0f / (1.0f + __expf(-v));
      Cm[idx] = g * Cm[idx] + (1.0f - g) * Xres[idx];
    } else {
      Cm[idx] = v;
    }
  }
}

// ---------------------------------------------------------------------------
// Flash attention: retrieved = softmax(X X^T) X.  keys == values == queries.
// Block: 128 threads = 4 waves; each wave owns 16 query rows.
// Q (64x512) and the 16-key block staged in dynamic LDS; O lives in VGPRs.
// ---------------------------------------------------------------------------
static constexpr int QSTR = 516;  // LDS row strides (bank-skew, 8B aligned)
static constexpr int KSTR = 516;
static constexpr int PSTR = 18;

__global__ __launch_bounds__(128)
void flash_attn(const float* __restrict__ X, float* __restrict__ Ret) {
  extern __shared__ float smem[];
  float* qs = smem;                  // 64 x QSTR
  float* ks = qs + 64 * QSTR;        // 16 x KSTR
  float* pb = ks + 16 * KSTR;        // 4 waves x 16 x PSTR (P transpose bounce)

  const int tid  = threadIdx.x;
  const int lane = tid & 31;
  const int wave = tid >> 5;
  const int ml = lane & 15;
  const int kh = lane >> 4;
  const int q0 = blockIdx.x * 64;

  // stage Q tile 64x512
  for (int i = tid; i < 64 * 128; i += 128) {
    int r = i >> 7, c = (i & 127) << 2;
    *(float4*)&qs[r * QSTR + c] = *(const float4*)&X[(size_t)(q0 + r) * kD + c];
  }

  v8f o[32];
  const v8f vzero = {0.f, 0.f, 0.f, 0.f, 0.f, 0.f, 0.f, 0.f};
#pragma unroll
  for (int t = 0; t < 32; ++t) o[t] = vzero;
  float mrun[8], lrun[8];
#pragma unroll
  for (int r = 0; r < 8; ++r) { mrun[r] = -INFINITY; lrun[r] = 0.0f; }

  const float* myQ  = &qs[(wave * 16 + ml) * QSTR];
  float*       myPB = &pb[wave * 16 * PSTR];

  for (int kb = 0; kb < kB; kb += 16) {
    __syncthreads();  // protect ks reuse from previous iteration
    for (int i = tid; i < 16 * 128; i += 128) {
      int r = i >> 7, c = (i & 127) << 2;
      *(float4*)&ks[r * KSTR + c] = *(const float4*)&X[(size_t)(kb + r) * kD + c];
    }
    __syncthreads();
    if (kb + 16 < kB)  // -> global_prefetch_b8 for the next key block
      __builtin_prefetch(&X[(size_t)(kb + 16) * kD + (size_t)tid * 16], 0, 1);

    // logits S(16x16) = Q(16x512) * Kblk(16x512)^T, 128 fp32 WMMAs
    v8f s = vzero;
#pragma unroll 8
    for (int kk = 0; kk < kD; kk += 4) {
      v2f a = *(const v2f*)&myQ[kk + kh * 2];
      v2f b = *(const v2f*)&ks[ml * KSTR + kk + kh * 2];
      s = wmma4(a, b, s);
    }

    // online softmax (row reductions across the 16 lanes holding each row)
    float corr[8];
#pragma unroll
    for (int r = 0; r < 8; ++r) {
      float sv = s[r];
      float rm = sv;
      rm = fmaxf(rm, __shfl_xor(rm, 1, 16));
      rm = fmaxf(rm, __shfl_xor(rm, 2, 16));
      rm = fmaxf(rm, __shfl_xor(rm, 4, 16));
      rm = fmaxf(rm, __shfl_xor(rm, 8, 16));
      float mn = fmaxf(mrun[r], rm);
      float c0 = __expf(mrun[r] - mn);  // 0 on first block (exp(-inf))
      float pe = __expf(sv - mn);
      float rs = pe;
      rs += __shfl_xor(rs, 1, 16);
      rs += __shfl_xor(rs, 2, 16);
      rs += __shfl_xor(rs, 4, 16);
      rs += __shfl_xor(rs, 8, 16);
      lrun[r] = lrun[r] * c0 + rs;
      mrun[r] = mn;
      corr[r] = c0;
      pb[(wave * 16 + kh * 8 + r) * PSTR + ml] = pe;  // bounce for transpose
    }
#pragma unroll
    for (int t = 0; t < 32; ++t) {
#pragma unroll
      for (int r = 0; r < 8; ++r) o[t][r] *= corr[r];
    }

    // O(16x512) += P(16x16) * Vblk(16x512): 4 K-chunks x 32 N-tiles of WMMA
#pragma unroll
    for (int kc = 0; kc < 4; ++kc) {
      v2f pa = *(const v2f*)&myPB[ml * PSTR + kc * 4 + kh * 2];
      const int kA = kc * 4 + kh * 2;
#pragma unroll
      for (int t = 0; t < 32; ++t) {
        v2f b;
        b.x = ks[kA * KSTR + t * 16 + ml];
        b.y = ks[(kA + 1) * KSTR + t * 16 + ml];
        o[t] = wmma4(pa, b, o[t]);
      }
    }
  }

  // normalize and store retrieved
#pragma unroll
  for (int r = 0; r < 8; ++r) {
    float inv = 1.0f / lrun[r];
    int row = q0 + wave * 16 + kh * 8 + r;
#pragma unroll
    for (int t = 0; t < 32; ++t)
      Ret[(size_t)row * kD + t * 16 + ml] = o[t][r] * inv;
  }
}

__global__ void fill_ones(float* __restrict__ p, int n) {
  int i = blockIdx.x * 256 + threadIdx.x;
  if (i < n) p[i] = 1.0f;
}

// ---------------------------------------------------------------------------
extern "C" void kernel_launch(void* const* d_in, const int* in_sizes, int n_in,
                              void* d_out, int out_size, void* d_ws,
                              size_t ws_size, hipStream_t stream) {
  (void)in_sizes; (void)n_in; (void)out_size; (void)d_ws; (void)ws_size;
  const float* X  = (const float*)d_in[0];
  // d_in[1] (memory_buffer) is fully overwritten by the scatter -> unused.
  const float* Wv = (const float*)d_in[2] + (size_t)2 * kD * kD;
  const float* bv = (const float*)d_in[3] + 2 * kD;
  const float* Wo = (const float*)d_in[4];
  const float* bo = (const float*)d_in[5];
  const float* Wg = (const float*)d_in[6];
  const float* bg = (const float*)d_in[7];

  float* gated = (float*)d_out;                 // [B, D]
  float* retr  = gated + (size_t)kB * kD;       // [B, D]
  float* attnw = retr + (size_t)kB * kD;        // [B] (== [B,1,1])
  float* tmpV  = retr;  // retrieved slot doubles as scratch for V projection

  dim3 gblk(256), ggrid(kD / 64, kB / 32);
  // 1) V = X*Wv^T + bv                  (temp in retrieved slot)
  gemm_nt_bias<<<ggrid, gblk, 0, stream>>>(X, Wv, bv, tmpV, kD, kD, 0, nullptr);
  // 2) attended = V*Wo^T + bo           (into gated slot)
  gemm_nt_bias<<<ggrid, gblk, 0, stream>>>(tmpV, Wo, bo, gated, kD, kD, 0, nullptr);
  // 3) gated = sigmoid(X*Wg^T+bg)*attended + (1-s)*X   (fused, in place)
  gemm_nt_bias<<<ggrid, gblk, 0, stream>>>(X, Wg, bg, gated, kD, kD, 1, X);
  // 4) retrieved = softmax(X X^T) X     (overwrites tmpV after it is consumed)
  size_t smem = (size_t)(64 * QSTR + 16 * KSTR + 4 * 16 * PSTR) * sizeof(float);
  hipError_t err = hipFuncSetAttribute(
      (const void*)flash_attn, hipFuncAttributeMaxDynamicSharedMemorySize,
      (int)smem);
  (void)err;
  flash_attn<<<dim3(kB / 64), dim3(128), smem, stream>>>(X, retr);
  // 5) attention weights are exactly 1 (softmax over a single key)
  fill_ones<<<dim3((kB + 255) / 256), dim3(256), 0, stream>>>(attnw, kB);
}